// NoiseCancellation_46918222742097
// MI455X (gfx1250) — compile-verified
//
#include <hip/hip_runtime.h>

#define NPTS   8192
#define NROW   64
#define NCOL   128
#define LDPAD  130     // 128 cols + 2 pad floats (bank-conflict-free column access, 8B aligned)
#define NCH    32
#define NBATCH 128
#define FORDER 10
#define PI_F   3.14159265358979323846f

typedef float v2f __attribute__((ext_vector_type(2)));
typedef float v8f __attribute__((ext_vector_type(8)));

// D = A(16x4) * B(4x16) + C, f32 WMMA. All modifier immediates must be 0 for the f32 variant
// (ISA: F32 NEG = {CNeg,0,0}); subtraction is done by negating the A fragment in VALU.
__device__ __forceinline__ v8f wmma4(v2f a, v2f b, v8f c) {
  return __builtin_amdgcn_wmma_f32_16x16x4_f32(false, a, false, b, (short)0, c, false, false);
}

// W_N^(i*j) from the shared W_8192 table; mult = 8192/N; sgn=-1 -> conjugate.
__device__ __forceinline__ void genW(const float* tab, int mult, int i, int j, float sgn,
                                     float& cc, float& ss) {
  int idx = (i * j * mult) & (NPTS - 1);
  cc = tab[2 * idx];
  ss = sgn * tab[2 * idx + 1];
}

// A-fragment (16x4 slab of a DFT matrix), ISA layout:
// lanes 0-15: row M=row0+lane, K=k0..k0+1 ; lanes 16-31: same rows, K=k0+2..k0+3
__device__ __forceinline__ void genAfrag(const float* tab, int mult, float sgn,
                                         int row0, int k0, int lane, v2f& ar, v2f& ai) {
  int m  = row0 + (lane & 15);
  int kk = k0 + ((lane >> 4) << 1);
  float c0, s0, c1, s1;
  genW(tab, mult, m, kk,     sgn, c0, s0);
  genW(tab, mult, m, kk + 1, sgn, c1, s1);
  ar.x = c0; ar.y = c1;
  ai.x = s0; ai.y = s1;
}

// B-fragment (4x16 slab of a DFT matrix): VGPR0 holds rows {k0, k0+2}, VGPR1 rows {k0+1, k0+3}
__device__ __forceinline__ void genBfrag(const float* tab, int mult, float sgn,
                                         int col0, int k0, int lane, v2f& br, v2f& bi) {
  int n  = col0 + (lane & 15);
  int kk = k0 + ((lane >> 4) << 1);
  float c0, s0, c1, s1;
  genW(tab, mult, kk,     n, sgn, c0, s0);
  genW(tab, mult, kk + 1, n, sgn, c1, s1);
  br.x = c0; br.y = c1;
  bi.x = s0; bi.y = s1;
}

__device__ __forceinline__ v2f ldsAfrag(const float* M, int row0, int k0, int lane) {
  int r  = row0 + (lane & 15);
  int kk = k0 + ((lane >> 4) << 1);
  v2f a;
  a.x = M[r * LDPAD + kk];
  a.y = M[r * LDPAD + kk + 1];
  return a;
}

__device__ __forceinline__ v2f ldsBfrag(const float* M, int col0, int k0, int lane) {
  int n  = col0 + (lane & 15);
  int kk = k0 + ((lane >> 4) << 1);
  v2f b;
  b.x = M[kk * LDPAD + n];
  b.y = M[(kk + 1) * LDPAD + n];
  return b;
}

// C/D 16x16 layout: VGPR v -> row (row0 + v + 8*(lane>>4)), col (col0 + lane&15)
__device__ __forceinline__ void stCfrag(float* M, int row0, int col0, int lane, v8f acc) {
  int half = lane >> 4;
  int col  = col0 + (lane & 15);
#pragma unroll
  for (int v = 0; v < 8; ++v) {
    M[(row0 + v + 8 * half) * LDPAD + col] = acc[v];
  }
}

// C(64x128) = DFT(gen, in A position) x B(LDS). B may be real-only; C may be real-only.
template <bool BCPLX, bool CCPLX>
__device__ void mm_dftA(const float* tab, int mult, float sgn,
                        const float* Br, const float* Bi,
                        float* Cr, float* Ci, int K, int wave, int lane) {
  for (int t = wave; t < 32; t += 8) {
    int row0 = (t >> 3) << 4;
    int col0 = (t & 7) << 4;
    v8f cr = {0.f, 0.f, 0.f, 0.f, 0.f, 0.f, 0.f, 0.f};
    v8f ci = {0.f, 0.f, 0.f, 0.f, 0.f, 0.f, 0.f, 0.f};
    for (int k0 = 0; k0 < K; k0 += 4) {
      v2f ar, ai;
      genAfrag(tab, mult, sgn, row0, k0, lane, ar, ai);
      v2f br = ldsBfrag(Br, col0, k0, lane);
      cr = wmma4(ar, br, cr);
      if (CCPLX) ci = wmma4(ai, br, ci);
      if (BCPLX) {
        v2f bi  = ldsBfrag(Bi, col0, k0, lane);
        v2f nai = -ai;                           // VALU negate (f32 WMMA has no A-neg)
        cr = wmma4(nai, bi, cr);                 // Cr -= Ai*Bi
        if (CCPLX) ci = wmma4(ar, bi, ci);
      }
    }
    stCfrag(Cr, row0, col0, lane, cr);
    if (CCPLX) stCfrag(Ci, row0, col0, lane, ci);
  }
}

// C(64x128) = A(LDS, complex) x DFT(gen, in B position)
__device__ void mm_dftB(const float* tab, int mult, float sgn,
                        const float* Ar, const float* Ai,
                        float* Cr, float* Ci, int K, int wave, int lane) {
  for (int t = wave; t < 32; t += 8) {
    int row0 = (t >> 3) << 4;
    int col0 = (t & 7) << 4;
    v8f cr = {0.f, 0.f, 0.f, 0.f, 0.f, 0.f, 0.f, 0.f};
    v8f ci = {0.f, 0.f, 0.f, 0.f, 0.f, 0.f, 0.f, 0.f};
    for (int k0 = 0; k0 < K; k0 += 4) {
      v2f ar = ldsAfrag(Ar, row0, k0, lane);
      v2f ai = ldsAfrag(Ai, row0, k0, lane);
      v2f br, bi;
      genBfrag(tab, mult, sgn, col0, k0, lane, br, bi);
      v2f nai = -ai;                             // VALU negate (f32 WMMA has no A-neg)
      cr = wmma4(ar, br, cr);
      cr = wmma4(nai, bi, cr);                   // Cr -= Ai*Bi
      ci = wmma4(ar, bi, ci);
      ci = wmma4(ai, br, ci);
    }
    stCfrag(Cr, row0, col0, lane, cr);
    stCfrag(Ci, row0, col0, lane, ci);
  }
}

// One workgroup per (batch, channel) chain. Four-step FFT -> threshold -> inverse -> ref.
__global__ __launch_bounds__(256) void nc_fft_denoise_kernel(const float* __restrict__ x,
                                                             float* __restrict__ refout) {
  extern __shared__ float smem[];
  float* Xr  = smem;
  float* Xi  = Xr + NROW * LDPAD;
  float* Yr  = Xi + NROW * LDPAD;
  float* Yi  = Yr + NROW * LDPAD;
  float* tab = Yi + NROW * LDPAD;   // 2*8192 floats: cos/sin of exp(-2*pi*i*k/8192)
  float* red = tab + 2 * NPTS;      // 256 floats

  const int tid  = threadIdx.x;
  const int wave = tid >> 5;
  const int lane = tid & 31;
  const int b = blockIdx.x >> 5;    // /32
  const int c = blockIdx.x & 31;
  const float* xp = x + (size_t)b * NPTS * NCH + c;
  float* rp = refout + (size_t)b * NPTS * NCH + c;

  // Build twiddle table: tab[k] = W_8192^k = exp(-2*pi*i*k/8192)
  for (int i = tid; i < NPTS; i += 256) {
    float s, cc;
    sincosf((-2.0f * PI_F / (float)NPTS) * (float)i, &s, &cc);
    tab[2 * i]     = cc;
    tab[2 * i + 1] = s;
  }
  // Load time series as 64x128 row-major matrix (t = 128*n1 + n2)
  for (int i = tid; i < NPTS; i += 256) {
    int p = (i >> 7) * LDPAD + (i & 127);
    Xr[p] = xp[(size_t)i * NCH];
    Xi[p] = 0.0f;
  }
  __syncthreads();

  // Step 1: Y[k1][n2] = sum_n1 W64^(n1*k1) * X[n1][n2]   (B real)
  mm_dftA<false, true>(tab, 128, 1.0f, Xr, nullptr, Yr, Yi, 64, wave, lane);
  __syncthreads();

  // Step 2: twiddle Y[k1][n2] *= W8192^(n2*k1)
  for (int i = tid; i < NPTS; i += 256) {
    int k1 = i >> 7, n2 = i & 127;
    int idx = k1 * n2;                       // < 8192
    float cc = tab[2 * idx], ss = tab[2 * idx + 1];
    int p = k1 * LDPAD + n2;
    float yr = Yr[p], yi = Yi[p];
    Yr[p] = yr * cc - yi * ss;
    Yi[p] = yr * ss + yi * cc;
  }
  __syncthreads();

  // Step 3: Z[k1][k2] = sum_n2 Y[k1][n2] * W128^(n2*k2)  -> into X planes
  mm_dftB(tab, 64, 1.0f, Yr, Yi, Xr, Xi, 128, wave, lane);
  __syncthreads();

  // Threshold: zero bins with |Z| < 0.1 * max|Z|  (compare squared magnitudes)
  float lm = 0.0f;
  for (int i = tid; i < NPTS; i += 256) {
    int p = (i >> 7) * LDPAD + (i & 127);
    lm = fmaxf(lm, Xr[p] * Xr[p] + Xi[p] * Xi[p]);
  }
  red[tid] = lm;
  __syncthreads();
  for (int s2 = 128; s2 > 0; s2 >>= 1) {
    if (tid < s2) red[tid] = fmaxf(red[tid], red[tid + s2]);
    __syncthreads();
  }
  float thr2 = 0.01f * red[0];   // (0.1*max)^2
  __syncthreads();
  for (int i = tid; i < NPTS; i += 256) {
    int p = (i >> 7) * LDPAD + (i & 127);
    float m2 = Xr[p] * Xr[p] + Xi[p] * Xi[p];
    if (m2 < thr2) { Xr[p] = 0.0f; Xi[p] = 0.0f; }
  }
  __syncthreads();

  // Inverse step A: D[k1][n2] = sum_k2 Z[k1][k2] * conj(W128)^(n2*k2) -> Y
  mm_dftB(tab, 64, -1.0f, Xr, Xi, Yr, Yi, 128, wave, lane);
  __syncthreads();

  // Inverse step B: D[k1][n2] *= conj(W8192)^(n2*k1)
  for (int i = tid; i < NPTS; i += 256) {
    int k1 = i >> 7, n2 = i & 127;
    int idx = k1 * n2;
    float cc = tab[2 * idx], ss = -tab[2 * idx + 1];
    int p = k1 * LDPAD + n2;
    float yr = Yr[p], yi = Yi[p];
    Yr[p] = yr * cc - yi * ss;
    Yi[p] = yr * ss + yi * cc;
  }
  __syncthreads();

  // Inverse step C: E[n1][n2] = sum_k1 conj(W64)^(n1*k1) * D[k1][n2]; real part only -> Xr
  mm_dftA<true, false>(tab, 128, -1.0f, Yr, Yi, Xr, nullptr, 64, wave, lane);
  __syncthreads();

  // ref = x - real(ifft)  (ifft includes 1/N)
  for (int i = tid; i < NPTS; i += 256) {
    int p = (i >> 7) * LDPAD + (i & 127);
    rp[(size_t)i * NCH] = xp[(size_t)i * NCH] - Xr[p] * (1.0f / (float)NPTS);
  }
}

// Sequential LMS, one lane per (b,c) chain; transforms d_out (holding ref) in place.
__global__ __launch_bounds__(32) void nc_lms_kernel(const float* __restrict__ x,
                                                    const float* __restrict__ w0,
                                                    float* __restrict__ out) {
  const int b = blockIdx.x;
  const int c = threadIdx.x;   // lane == channel -> coalesced
  const float* xp = x + (size_t)b * NPTS * NCH + c;
  float* op = out + (size_t)b * NPTS * NCH + c;

  float w[FORDER], r[FORDER];
#pragma unroll
  for (int j = 0; j < FORDER; ++j) w[j] = w0[((size_t)b * FORDER + j) * NCH + c];
  // window r[j] = ref[n-1-j]; initialize for n = FORDER from ref[0..9] (currently in out)
#pragma unroll
  for (int j = 0; j < FORDER; ++j) r[j] = op[(size_t)(FORDER - 1 - j) * NCH];
  // first FORDER outputs are passthrough of x
#pragma unroll
  for (int n = 0; n < FORDER; ++n) op[(size_t)n * NCH] = xp[(size_t)n * NCH];

  for (int base = FORDER; base < NPTS; base += FORDER) {
#pragma unroll
    for (int u = 0; u < FORDER; ++u) {
      int n = base + u;
      if (n < NPTS) {
        float rn = op[(size_t)n * NCH];          // ref[n], read before overwrite
        float y = (((w[0] * r[0] + w[1] * r[1]) + (w[2] * r[2] + w[3] * r[3])) +
                   ((w[4] * r[4] + w[5] * r[5]) + (w[6] * r[6] + w[7] * r[7]))) +
                  (w[8] * r[8] + w[9] * r[9]);
        float e = xp[(size_t)n * NCH] - y;
        float g = 0.02f * e;                     // 2*MU*e
#pragma unroll
        for (int j = 0; j < FORDER; ++j) w[j] = fmaf(g, r[j], w[j]);
        op[(size_t)n * NCH] = e;
#pragma unroll
        for (int j = FORDER - 1; j > 0; --j) r[j] = r[j - 1];
        r[0] = rn;
      }
    }
  }
}

extern "C" void kernel_launch(void* const* d_in, const int* in_sizes, int n_in,
                              void* d_out, int out_size, void* d_ws, size_t ws_size,
                              hipStream_t stream) {
  const float* x = (const float*)d_in[0];
  const float* w = (const float*)d_in[1];
  float* out = (float*)d_out;
  (void)d_ws; (void)ws_size; (void)in_sizes; (void)n_in; (void)out_size;

  size_t smem = (size_t)(4 * NROW * LDPAD + 2 * NPTS + 256) * sizeof(float); // ~195 KB
  nc_fft_denoise_kernel<<<dim3(NBATCH * NCH), dim3(256), smem, stream>>>(x, out);
  nc_lms_kernel<<<dim3(NBATCH), dim3(32), 0, stream>>>(x, w, out);
}